// CannyNet_86071144612474
// MI455X (gfx1250) — compile-verified
//
#include <hip/hip_runtime.h>
#include <cstdint>
#include <cstddef>

#define IH 2048
#define IW 2048
#define TS 32           // output tile
#define NTHREADS 256    // 8 waves of 32 (wave32)

typedef float v2f __attribute__((ext_vector_type(2)));
typedef float v8f __attribute__((ext_vector_type(8)));

#if __has_builtin(__builtin_amdgcn_wmma_f32_16x16x4_f32)
#define HAVE_WMMA_F32X4 1
#endif

// LDS float offsets (s_in row stride padded to 44 for bank-conflict-free
// strided fragment reads while keeping 16B-aligned rows for async B128).
#define SIN_STRIDE 44
#define SIN_OFF    0        // 3*40*44 = 5280
#define SHB_OFF    5280     // 3*40*36 = 4320
#define SBL_OFF    9600     // 3*36*36 = 3888
#define DUMP_OFF   13488    // 32 per-lane dump slots for branch-free stores
#define SMEM_FLOATS 13520   // 54,080 bytes

// ---------------------------------------------------------------------------
// Fused Canny pipeline, one 32x32 tile per workgroup.
//   A: img tile + halo4 (3 x 40 x 40) -> LDS   (async DMA on interior tiles)
//   B: horizontal 5-tap gauss via v_wmma_f32_16x16x4_f32 (data x band)
//   C: vertical   5-tap gauss via v_wmma_f32_16x16x4_f32 (band x data)
//   D: sobel + per-channel magnitude -> s_mag/s_gx/s_gy; emits blurred_img
//   E: orientation quantize + NMS + thresholds, emits 5 planes
// ---------------------------------------------------------------------------
__global__ __launch_bounds__(NTHREADS) void canny_fused(
    const float* __restrict__ img,   // [3, H, W]
    const float* __restrict__ gauss, // [5]
    const float* __restrict__ thr_p, // [1]
    float* __restrict__ out)         // 8 * H * W, concatenated planes
{
  __shared__ __align__(16) float smem[SMEM_FLOATS];
  float* const s_in  = smem + SIN_OFF;
  float* const s_hb  = smem + SHB_OFF;
  float* const s_bl  = smem + SBL_OFF;
  float* const s_mag = smem;          // 34*34 = 1156, overlays s_in (dead)
  float* const s_gx  = smem + 1156;
  float* const s_gy  = smem + 2312;

  const int tid  = threadIdx.x;
  const int tx0  = blockIdx.x * TS;
  const int ty0  = blockIdx.y * TS;
  const size_t HW = (size_t)IH * IW;

  const float g0 = gauss[0], g1 = gauss[1], g2 = gauss[2],
              g3 = gauss[3], g4 = gauss[4];
  const float thr = thr_p[0];

  const int wid  = tid >> 5;          // wave id (8 waves)
  const int lane = tid & 31;
  const int half = lane >> 4;         // K-half select (WMMA fragment layout)
  const int lm   = lane & 15;

  // ---------------- stage A: input tile + halo(4) into LDS ----------------
  const bool interior = (tx0 >= 4) & (ty0 >= 4) &
                        (tx0 + 36 <= IW) & (ty0 + 36 <= IH);
  if (interior) {
    // Async DMA: 3 ch * 40 rows * 10 x 16B chunks = 1200 lane-transfers.
    const uint32_t lds_base = (uint32_t)(uintptr_t)(const void*)s_in;
    for (int c = tid; c < 1200; c += NTHREADS) {
      const int ch  = c / 400;
      const int rem = c - ch * 400;
      const int row = rem / 10;
      const int c16 = rem - row * 10;
      const float* gp = img + (size_t)ch * HW
                            + (size_t)(ty0 - 4 + row) * IW
                            + (size_t)(tx0 - 4 + c16 * 4);          // 16B aligned
      const uint32_t loff = lds_base +
          (uint32_t)(((ch * 40 + row) * SIN_STRIDE + c16 * 4) * 4); // 16B aligned
      asm volatile("global_load_async_to_lds_b128 %0, %1, off"
                   :: "v"(loff), "v"(gp) : "memory");
    }
#if __has_builtin(__builtin_amdgcn_s_wait_asynccnt)
    __builtin_amdgcn_s_wait_asynccnt(0);
#else
    asm volatile("s_wait_asynccnt 0" ::: "memory");
#endif
  } else {
    for (int i = tid; i < 4800; i += NTHREADS) {
      const int ch  = i / 1600;
      const int rem = i - ch * 1600;
      const int r   = rem / 40;
      const int c   = rem - r * 40;
      const int x = tx0 - 4 + c, y = ty0 - 4 + r;
      float v = 0.0f;
      if ((unsigned)x < (unsigned)IW && (unsigned)y < (unsigned)IH)
        v = img[(size_t)ch * HW + (size_t)y * IW + (size_t)x];
      s_in[(ch * 40 + r) * SIN_STRIDE + c] = v;
    }
  }
  __syncthreads();

#ifdef HAVE_WMMA_F32X4
  // Band-matrix fragment constants, band[d] = g[d] for d in [0,5).
  // Identical per-lane values serve as B (h-blur) and A (v-blur):
  //   value(q, j) = band(4q + 2*half + j - lane%16)
  float cst[5][2];
  #pragma unroll
  for (int q = 0; q < 5; ++q) {
    #pragma unroll
    for (int j = 0; j < 2; ++j) {
      const int d = 4 * q + 2 * half + j - lm;
      cst[q][j] = (d == 0) ? g0 : (d == 1) ? g1 : (d == 2) ? g2
                : (d == 3) ? g3 : (d == 4) ? g4 : 0.0f;
    }
  }

  // ------------- stage B: horizontal gaussian via WMMA (A=data,B=band) ------
  // s_hb[ch][r][c] (40x36): out = s_in_block(16x20) x band(20x16).
  // Valid outputs only touch valid inputs (banded); padded rows/cols go to a
  // per-lane dump slot via address select (no EXEC toggling around stores).
  for (int blk = wid; blk < 27; blk += 8) {          // 3ch x 3br x 3bc, per-wave
    const int ch = blk / 9;
    const int r9 = blk - ch * 9;
    const int br = r9 / 3;
    const int bc = r9 - br * 3;
    v8f acc = {};
    #pragma unroll
    for (int q = 0; q < 5; ++q) {
      const int ai = (ch * 40 + br * 16 + lm) * SIN_STRIDE
                   + bc * 16 + 4 * q + 2 * half;
      v2f a; a.x = s_in[ai]; a.y = s_in[ai + 1];
      v2f b; b.x = cst[q][0]; b.y = cst[q][1];
      acc = __builtin_amdgcn_wmma_f32_16x16x4_f32(false, a, false, b,
                                                  (short)0, acc, false, false);
    }
    const int col = bc * 16 + lm;
    const bool colOK = (col < 36);
    #pragma unroll
    for (int r = 0; r < 8; ++r) {
      const int row = br * 16 + 8 * half + r;
      const bool ok = colOK & (row < 40);
      const int idx = ok ? (SHB_OFF + (ch * 40 + row) * 36 + col)
                         : (DUMP_OFF + lane);
      smem[idx] = acc[r];
    }
  }
#else
  for (int i = tid; i < 4320; i += NTHREADS) {
    const int ch  = i / 1440;
    const int rem = i - ch * 1440;
    const int r   = rem / 36;
    const int c   = rem - r * 36;
    const float* p = &s_in[(ch * 40 + r) * SIN_STRIDE + c];
    s_hb[(ch * 40 + r) * 36 + c] =
        g0 * p[0] + g1 * p[1] + g2 * p[2] + g3 * p[3] + g4 * p[4];
  }
#endif
  __syncthreads();

  // ------------- stage C: vertical gaussian via WMMA (A=band,B=data) --------
  // s_bl[ch][r][c] (36x36): (r,c) <-> (y,x) = (ty0-2+r, tx0-2+c).
  // Out-of-image coords forced to 0 (sobel zero-pads blurred).
#ifdef HAVE_WMMA_F32X4
  for (int blk = wid; blk < 27; blk += 8) {
    const int ch = blk / 9;
    const int r9 = blk - ch * 9;
    const int br = r9 / 3;
    const int bc = r9 - br * 3;
    v8f acc = {};
    #pragma unroll
    for (int q = 0; q < 5; ++q) {
      const int bi = (ch * 40 + br * 16 + 4 * q + 2 * half) * 36
                   + bc * 16 + lm;
      v2f b; b.x = s_hb[bi]; b.y = s_hb[bi + 36];
      v2f a; a.x = cst[q][0]; a.y = cst[q][1];
      acc = __builtin_amdgcn_wmma_f32_16x16x4_f32(false, a, false, b,
                                                  (short)0, acc, false, false);
    }
    const int col = bc * 16 + lm;
    const bool colOK = (col < 36);
    const int x = tx0 - 2 + col;
    const bool xOK = ((unsigned)x < (unsigned)IW);
    #pragma unroll
    for (int r = 0; r < 8; ++r) {
      const int row = br * 16 + 8 * half + r;
      const int y = ty0 - 2 + row;
      float v = acc[r];
      if (!(xOK & ((unsigned)y < (unsigned)IH))) v = 0.0f;
      const bool ok = colOK & (row < 36);
      const int idx = ok ? (SBL_OFF + (ch * 36 + row) * 36 + col)
                         : (DUMP_OFF + lane);
      smem[idx] = v;
    }
  }
#else
  for (int i = tid; i < 3888; i += NTHREADS) {
    const int ch  = i / 1296;
    const int rem = i - ch * 1296;
    const int r   = rem / 36;
    const int c   = rem - r * 36;
    const int y = ty0 - 2 + r, x = tx0 - 2 + c;
    const float* p = &s_hb[(ch * 40 + r) * 36 + c];
    float v = g0 * p[0] + g1 * p[36] + g2 * p[72] + g3 * p[108] + g4 * p[144];
    if (!(((unsigned)x < (unsigned)IW) & ((unsigned)y < (unsigned)IH)))
      v = 0.0f;
    s_bl[(ch * 36 + r) * 36 + c] = v;
  }
#endif
  __syncthreads();

  // ---------------- stage D: sobel + magnitude, emit blurred_img -----------
  // s_mag/s_gx/s_gy on 34x34 : (r,c) <-> (y,x) = (ty0-1+r, tx0-1+c)
  // mag forced to 0 outside the image (dir-filter conv zero-pads grad_mag).
  // b[37] is the blurred pixel at (y,x): store central 32x32 to output here.
  for (int i = tid; i < 1156; i += NTHREADS) {
    const int r = i / 34;
    const int c = i - r * 34;
    const int y = ty0 - 1 + r, x = tx0 - 1 + c;
    float mag = 0.0f, gxs = 0.0f, gys = 0.0f;
    if ((unsigned)y < (unsigned)IH && (unsigned)x < (unsigned)IW) {
      const bool central = (r >= 1) & (r < 33) & (c >= 1) & (c < 33);
      #pragma unroll
      for (int ch = 0; ch < 3; ++ch) {
        const float* b = &s_bl[(ch * 36 + r) * 36 + c];
        const float a00 = b[0],  a01 = b[1],  a02 = b[2];
        const float a10 = b[36], a11 = b[37], a12 = b[38];
        const float a20 = b[72], a21 = b[73], a22 = b[74];
        // sobel_h = [[1,0,-1],[2,0,-2],[1,0,-1]], sobel_v = transpose
        const float gx = (a00 - a02) + 2.0f * (a10 - a12) + (a20 - a22);
        const float gy = (a00 - a20) + 2.0f * (a01 - a21) + (a02 - a22);
        mag += sqrtf(gx * gx + gy * gy);
        gxs += gx;
        gys += gy;
        if (central)
          out[(size_t)ch * HW + (size_t)y * IW + (size_t)x] = a11;
      }
    }
    s_mag[r * 34 + c] = mag;
    s_gx [r * 34 + c] = gxs;
    s_gy [r * 34 + c] = gys;
  }
  __syncthreads();

  // ---------------- stage E: orientation, NMS, thresholds ------------------
  const float DEG = (float)(180.0 / 3.14159);   // reference's pi approximation
  for (int i = tid; i < TS * TS; i += NTHREADS) {
    const int rr = i >> 5, cc = i & 31;
    const int mr = rr + 1, mc = cc + 1;
    const float mag = s_mag[mr * 34 + mc];
    const float gx  = s_gx [mr * 34 + mc];
    const float gy  = s_gy [mr * 34 + mc];

    const float deg = atan2f(gy, gx) * DEG;
    const float q   = rintf((deg + 180.0f) / 45.0f); // half-to-even == jnp.round
    const float orient = 45.0f * q;
    const int   p = ((int)q) & 7;                    // q in [0,8], 8 -> 0

    // dx:+1 for p in {0,1,7}, -1 for {3,4,5}; dy:+1 for {1,2,3}, -1 for {5,6,7}
    const int dx = (int)((0x83u >> p) & 1u) - (int)((0x38u >> p) & 1u);
    const int dy = (int)((0x0Eu >> p) & 1u) - (int)((0xE0u >> p) & 1u);

    const float pos = mag - s_mag[(mr + dy) * 34 + (mc + dx)];
    const float neg = mag - s_mag[(mr - dy) * 34 + (mc - dx)];

    const float thin  = (fminf(pos, neg) > 0.0f) ? mag : 0.0f;
    const float thrd  = (thin < thr) ? 0.0f : thin;
    const float early = (mag  < thr) ? 0.0f : mag;

    const size_t o = (size_t)(ty0 + rr) * IW + (size_t)(tx0 + cc);
    out[3 * HW + o] = mag;
    out[4 * HW + o] = orient;
    out[5 * HW + o] = thin;
    out[6 * HW + o] = thrd;
    out[7 * HW + o] = early;
  }
}

extern "C" void kernel_launch(void* const* d_in, const int* in_sizes, int n_in,
                              void* d_out, int out_size, void* d_ws, size_t ws_size,
                              hipStream_t stream) {
  (void)in_sizes; (void)n_in; (void)out_size; (void)d_ws; (void)ws_size;
  const float* img   = (const float*)d_in[0];  // [1,3,2048,2048]
  const float* thr   = (const float*)d_in[1];  // [1]
  const float* gauss = (const float*)d_in[2];  // gauss_h taps [5]
  dim3 grid(IW / TS, IH / TS);                 // 64 x 64 tiles
  canny_fused<<<grid, NTHREADS, 0, stream>>>(img, gauss, thr, (float*)d_out);
}